// GridEncoder_37744172597967
// MI455X (gfx1250) — compile-verified
//
#include <hip/hip_runtime.h>
#include <hip/hip_bf16.h>

// Problem constants (B=1 fixed by the reference)
#define Wd 96
#define Hd 96
#define Cd 64
#define Dd 48
#define Vd 3
#define HWd (Hd * Wd)
#define DHWd (Dd * HWd)

typedef __attribute__((ext_vector_type(2))) float v2f;
typedef __attribute__((ext_vector_type(8))) float v8f;

// ---------------------------------------------------------------------------
// Kernel 1: per-pixel projective transform Rref = R @ [x, y, 1] done on the
// CDNA5 matrix pipe.  D(16x16) = A(16x4) x B(4x16):
//   A rows 0..2 = rows of R (K=0..2 = R columns, K=3 = 0)
//   B column n  = [x_n, y_n, 1, 0]
// VGPR layouts per CDNA5 ISA 7.12.2 (32-bit A 16x4, 32-bit C/D 16x16).
// Output: rref[(view-1)*3*HW + comp*HW + pix]
// ---------------------------------------------------------------------------
__global__ __launch_bounds__(32) void rref_wmma_kernel(
    const float* __restrict__ proj,   // (V,3,4)
    float* __restrict__ rref)         // (V-1, 3, HW)
{
    const int lane = threadIdx.x & 31;
    const int m    = lane & 15;
    const bool hiK = lane >= 16;
    const int view = blockIdx.y + 1;          // source views 1..V-1
    const int pixBase = blockIdx.x * 16;      // 16 pixels per wave

    const float* Pv = proj + view * 12;       // row-major 3x4

    // A operand: lane<16 -> {R[m][0], R[m][1]}, lane>=16 -> {R[m][2], 0}
    v2f a;
    float a0 = 0.f, a1 = 0.f;
    if (m < 3) {
        if (!hiK) { a0 = Pv[m * 4 + 0]; a1 = Pv[m * 4 + 1]; }
        else      { a0 = Pv[m * 4 + 2]; a1 = 0.f; }
    }
    a.x = a0; a.y = a1;

    // B operand: column n = [x, y, 1, 0]
    const int pix = pixBase + m;
    const float xf = (float)(pix % Wd);
    const float yf = (float)(pix / Wd);
    v2f b;
    if (!hiK) { b.x = xf;  b.y = yf;  }   // K=0 (x), K=1 (y)
    else      { b.x = 1.f; b.y = 0.f; }   // K=2 (1), K=3 (0)

    v8f c = {};
    c = __builtin_amdgcn_wmma_f32_16x16x4_f32(
        /*neg_a=*/false, a, /*neg_b=*/false, b,
        /*c_mod=*/(short)0, c, /*reuse_a=*/false, /*reuse_b=*/false);

    // D: lanes 0..15, VGPR r holds D[M=r][N=lane]; rows 0..2 are the result.
    if (lane < 16 && pix < HWd) {
        float* dst = rref + (view - 1) * 3 * HWd;
        dst[0 * HWd + pix] = c[0];
        dst[1 * HWd + pix] = c[1];
        dst[2 * HWd + pix] = c[2];
    }
}

// ---------------------------------------------------------------------------
// Per-view bilinear sampling state (grid_sample, padding=zeros,
// align_corners=True).  Tap validity is folded into the weights so the
// per-channel inner loop is 4 loads + 4 FMAs.
// ---------------------------------------------------------------------------
struct ViewSamp {
    int   i00, i01, i10, i11;
    float w00, w01, w10, w11;
    float m;
};

__device__ inline ViewSamp make_samp(float px, float py, float pz) {
    const float sx = px / pz;
    const float sy = py / pz;
    // normalize exactly as reference, then de-normalize as reference
    const float gx = sx / ((Wd - 1) * 0.5f) - 1.0f;
    const float gy = sy / ((Hd - 1) * 0.5f) - 1.0f;
    const float x = (gx + 1.0f) * 0.5f * (float)(Wd - 1);
    const float y = (gy + 1.0f) * 0.5f * (float)(Hd - 1);

    const float x0 = floorf(x);
    const float y0 = floorf(y);
    const float wx1 = x - x0, wy1 = y - y0;
    const float wx0 = 1.0f - wx1, wy0 = 1.0f - wy1;

    const float vx0 = (x0 >= 0.0f       && x0 <= (float)(Wd - 1)) ? 1.0f : 0.0f;
    const float vx1 = (x0 >= -1.0f      && x0 <= (float)(Wd - 2)) ? 1.0f : 0.0f;
    const float vy0 = (y0 >= 0.0f       && y0 <= (float)(Hd - 1)) ? 1.0f : 0.0f;
    const float vy1 = (y0 >= -1.0f      && y0 <= (float)(Hd - 2)) ? 1.0f : 0.0f;

    const int ix0 = (int)fminf(fmaxf(x0,        0.0f), (float)(Wd - 1));
    const int ix1 = (int)fminf(fmaxf(x0 + 1.0f, 0.0f), (float)(Wd - 1));
    const int iy0 = (int)fminf(fmaxf(y0,        0.0f), (float)(Hd - 1));
    const int iy1 = (int)fminf(fmaxf(y0 + 1.0f, 0.0f), (float)(Hd - 1));

    ViewSamp s;
    s.i00 = iy0 * Wd + ix0;  s.w00 = wx0 * wy0 * (vx0 * vy0);
    s.i01 = iy0 * Wd + ix1;  s.w01 = wx1 * wy0 * (vx1 * vy0);
    s.i10 = iy1 * Wd + ix0;  s.w10 = wx0 * wy1 * (vx0 * vy1);
    s.i11 = iy1 * Wd + ix1;  s.w11 = wx1 * wy1 * (vx1 * vy1);
    s.m = (gx > -1.0f && gx < 1.0f && gy > -1.0f && gy < 1.0f) ? 1.0f : 0.0f;
    return s;
}

// ---------------------------------------------------------------------------
// Kernel 2: one thread per voxel (d major, pixel minor -> coalesced stores,
// uniform depth per wave).  Warp params computed once, reused for 3 image
// channels per view and 64 variance channels.
// Output channels: [ref_img(3), wimg1(3), wimg2(3), var(64)]
// ---------------------------------------------------------------------------
__global__ __launch_bounds__(256) void fuse_kernel(
    const float* __restrict__ imgs,    // (V,3,H,W)
    const float* __restrict__ feats,   // (V,C,H,W)
    const float* __restrict__ proj,    // (V,3,4)
    const float* __restrict__ depth,   // (D)
    const float* __restrict__ rref,    // (V-1,3,HW) or nullptr
    float* __restrict__ out)           // (73,D,H,W)
{
    const int tid = blockIdx.x * blockDim.x + threadIdx.x;
    if (tid >= DHWd) return;
    const int pix = tid % HWd;   // h*W + w
    const int d   = tid / HWd;

    const float dep  = depth[d];
    const float invd = 1.0f / dep;

    ViewSamp s[2];
#pragma unroll
    for (int v = 0; v < 2; ++v) {
        const float* Pv = proj + (v + 1) * 12;
        float R0, R1, R2;
        if (rref != nullptr) {
            const float* rr = rref + v * 3 * HWd;
            R0 = rr[pix];
            R1 = rr[HWd + pix];
            R2 = rr[2 * HWd + pix];
        } else {
            const float xf = (float)(pix % Wd);
            const float yf = (float)(pix / Wd);
            R0 = Pv[0] * xf + Pv[1] * yf + Pv[2];
            R1 = Pv[4] * xf + Pv[5] * yf + Pv[6];
            R2 = Pv[8] * xf + Pv[9] * yf + Pv[10];
        }
        const float px = R0 + Pv[3]  * invd;
        const float py = R1 + Pv[7]  * invd;
        const float pz = R2 + Pv[11] * invd;
        s[v] = make_samp(px, py, pz);
    }

    // --- reference image broadcast (channels 0..2) ---
#pragma unroll
    for (int ch = 0; ch < 3; ++ch) {
        out[(ch * Dd + d) * HWd + pix] = imgs[ch * HWd + pix];
    }

    // --- warped source images (channels 3..8) ---
#pragma unroll
    for (int v = 0; v < 2; ++v) {
        const float* iv = imgs + (v + 1) * 3 * HWd;
#pragma unroll
        for (int ch = 0; ch < 3; ++ch) {
            const float* p = iv + ch * HWd;
            const float val = p[s[v].i00] * s[v].w00 + p[s[v].i01] * s[v].w01 +
                              p[s[v].i10] * s[v].w10 + p[s[v].i11] * s[v].w11;
            out[((3 + v * 3 + ch) * Dd + d) * HWd + pix] = val;
        }
    }

    // --- variance over C=64 feature channels (channels 9..72) ---
    const float cnt = 1.0f / (1.0f + s[0].m + s[1].m);
    const float* f0 = feats;                 // ref view
    const float* f1 = feats + Cd * HWd;      // view 1
    const float* f2 = feats + 2 * Cd * HWd;  // view 2
    float* ov = out + 9 * DHWd + d * HWd + pix;

#pragma unroll 4
    for (int c = 0; c < Cd; ++c) {
        const float r = f0[c * HWd + pix];
        const float* p1 = f1 + c * HWd;
        const float* p2 = f2 + c * HWd;
        const float a = p1[s[0].i00] * s[0].w00 + p1[s[0].i01] * s[0].w01 +
                        p1[s[0].i10] * s[0].w10 + p1[s[0].i11] * s[0].w11;
        const float b = p2[s[1].i00] * s[1].w00 + p2[s[1].i01] * s[1].w01 +
                        p2[s[1].i10] * s[1].w10 + p2[s[1].i11] * s[1].w11;
        const float sm = r + a + b;
        const float sq = r * r + a * a + b * b;
        const float mean = sm * cnt;
        ov[c * DHWd] = sq * cnt - mean * mean;
    }
}

extern "C" void kernel_launch(void* const* d_in, const int* in_sizes, int n_in,
                              void* d_out, int out_size, void* d_ws, size_t ws_size,
                              hipStream_t stream) {
    const float* imgs  = (const float*)d_in[0];  // (1,V,3,H,W)
    const float* feats = (const float*)d_in[1];  // (1,V,C,H,W)
    const float* proj  = (const float*)d_in[2];  // (1,V,3,4)
    const float* depth = (const float*)d_in[3];  // (1,D)
    float* out = (float*)d_out;                  // (1,73,D,H,W)

    // WMMA precompute of R @ [x,y,1] per pixel per source view into scratch.
    float* rref = nullptr;
    const size_t need = (size_t)(Vd - 1) * 3 * HWd * sizeof(float);
    if (ws_size >= need) {
        rref = (float*)d_ws;
        dim3 grid(HWd / 16, Vd - 1);
        rref_wmma_kernel<<<grid, 32, 0, stream>>>(proj, rref);
    }

    const int total = DHWd;
    fuse_kernel<<<(total + 255) / 256, 256, 0, stream>>>(
        imgs, feats, proj, depth, rref, out);
}